// LocalGatedResidualNetwork_42889543418070
// MI455X (gfx1250) — compile-verified
//
#include <hip/hip_runtime.h>
#include <math.h>

// ---- problem constants (from reference) ----
#define NB     16      // batch
#define NC     16      // channels
#define LIN    1534    // input length
#define KSZ    512     // kernel size
#define O1     1023    // layer-1 outputs
#define O2     512     // layer-2 outputs
#define LOUT   1535    // padded output length
#define EPSV   1e-3f

// ---- layer-1 WMMA tiling ----
#define RCH    16              // union-rows per chunk
#define NCHUNK 33              // 33*16 = 528 rows covers r in [0, 527]
#define KCH    (RCH * NC)      // 256 K-elements per chunk
#define PITCH  (KCH + 4)       // 260 ; 260 % 64 == 4 -> conflict-free fragment reads
#define KSPLIT 4               // K-chunks split across 4 blocks per tile
#define H1P    1024            // padded h1 row stride
#define STEPS_PER_WAVE 8       // (KCH/4) / 8 waves

typedef __attribute__((ext_vector_type(2))) float v2f;
typedef __attribute__((ext_vector_type(4))) float v4f;
typedef __attribute__((ext_vector_type(8))) float v8f;

// ---------------- kernel 0: zero the h1 accumulator ----------------
__global__ __launch_bounds__(256) void k0_zero(float* __restrict__ h1)
{
    int i = blockIdx.x * 256 + threadIdx.x;
    if (i < NB * H1P) h1[i] = 0.0f;
}

// ---------------- kernel 1: layer-1 locally-connected via WMMA f32 16x16x4 ----
// Banded GEMM per 16-output tile:
//   D[16b x 16j] += A[16b x K] * B[K x 16j],  K = (r,c), r in [0,528), c in [0,16)
//   A[b,(r,c)] = x[b, o0+r, c];  B[(r,c),j] = W1[o0+j, r-j, c] (0 outside band)
// grid = 64 tiles * KSPLIT blocks; 256 threads = 8 waves; each wave owns a
// contiguous block of 8 K-steps per chunk (literal trip counts, no exec masking).
__global__ __launch_bounds__(256) void k1_h1(const float* __restrict__ x,
                                             const float* __restrict__ W1,
                                             float* __restrict__ h1acc)
{
    __shared__ float Alds[NB * PITCH];   // x slab chunk, [b][kk]
    __shared__ float Blds[16 * PITCH];   // banded weights, [j][kk]
    __shared__ float Plds[8 * 256];      // per-wave partial D tiles

    const int tid   = threadIdx.x;
    const int tile  = blockIdx.x / KSPLIT;
    const int split = blockIdx.x % KSPLIT;
    const int o0    = tile * 16;
    const int wave  = tid >> 5;
    const int hi    = (tid >> 4) & 1;    // lane/16
    const int ln    = tid & 15;          // lane%16

    // fragment base: lane ln reads row ln of A (M=batch) / col ln of B (N=j);
    // lanes 16-31 take K+2,K+3 (f32 16x16x4 layout). Wave's steps are contiguous.
    const int fragbase = ln * PITCH + wave * (STEPS_PER_WAVE * 4) + 2 * hi;

    v8f acc = {};

    for (int ch = split; ch < NCHUNK; ch += KSPLIT) {
        const int r0 = ch * RCH;
        // ---- stage x chunk: A[b][kk] = x[b, o0+r0+kk/16, kk%16] (vec4) ----
        #pragma unroll
        for (int rep = 0; rep < (NB * KCH / 4) / 256; ++rep) {
            int i  = tid + rep * 256;
            int b  = i / (KCH / 4);
            int kk = (i % (KCH / 4)) * 4;
            int row = o0 + r0 + (kk >> 4);
            v4f v = {};
            if (row < LIN)
                v = *(const v4f*)(x + ((size_t)b * LIN + row) * NC + (kk & 15));
            *(v4f*)(Alds + b * PITCH + kk) = v;
        }
        // ---- stage banded W chunk: B[j][kk] = W1[o0+j, r0+kk/16 - j, kk%16] ----
        #pragma unroll
        for (int rep = 0; rep < (16 * KCH / 4) / 256; ++rep) {
            int i  = tid + rep * 256;
            int j  = i / (KCH / 4);
            int kk = (i % (KCH / 4)) * 4;
            int off = (r0 - j) * NC + kk;        // element offset within a W1 row
            int o   = o0 + j;
            v4f v = {};
            if (o < O1 && off >= 0 && off < KSZ * NC)
                v = *(const v4f*)(W1 + (size_t)o * (KSZ * NC) + off);
            *(v4f*)(Blds + j * PITCH + kk) = v;
        }
        __syncthreads();
        // ---- 8 WMMAs per wave: load all fragments, then multiply-accumulate ----
        v2f af[STEPS_PER_WAVE], bf[STEPS_PER_WAVE];
        #pragma unroll
        for (int it = 0; it < STEPS_PER_WAVE; ++it) {
            af[it] = *(const v2f*)(Alds + fragbase + it * 4);
            bf[it] = *(const v2f*)(Blds + fragbase + it * 4);
        }
        #pragma unroll
        for (int it = 0; it < STEPS_PER_WAVE; ++it)
            acc = __builtin_amdgcn_wmma_f32_16x16x4_f32(
                false, af[it], false, bf[it], (short)0, acc, false, false);
        __syncthreads();
    }

    // ---- cross-wave reduction; D layout: M = v + 8*hi (batch), N = ln (j) ----
    #pragma unroll
    for (int v = 0; v < 8; ++v)
        Plds[wave * 256 + (v + 8 * hi) * 16 + ln] = acc[v];
    __syncthreads();
    {
        int b = tid >> 4, j = tid & 15, o = o0 + j;
        float s = 0.0f;
        #pragma unroll
        for (int w = 0; w < 8; ++w) s += Plds[w * 256 + tid];
        if (o < O1) atomicAdd(&h1acc[(size_t)b * H1P + o], s);
    }
}

// ---------------- kernel 1b: bias + ELU in place ----------------
__global__ __launch_bounds__(256) void k1b_elu(float* __restrict__ h1,
                                               const float* __restrict__ b1)
{
    int i = blockIdx.x * 256 + threadIdx.x;
    if (i >= NB * O1) return;
    int b = i / O1, o = i % O1;
    float s = h1[(size_t)b * H1P + o] + b1[o];
    h1[(size_t)b * H1P + o] = (s > 0.0f) ? s : (expf(s) - 1.0f);
}

// ---------------- kernel 2: layer-2 locally-connected (C=1) ----------------
__global__ __launch_bounds__(256) void k2_h2(const float* __restrict__ h1,
                                             const float* __restrict__ W2,
                                             const float* __restrict__ b2,
                                             float* __restrict__ h2)
{
    __shared__ float red[256];
    int t = blockIdx.x, tid = threadIdx.x;
    int b = tid & 15, slice = tid >> 4;          // 16 slices x 32 taps
    float s = 0.0f;
    #pragma unroll 8
    for (int q = 0; q < 32; ++q) {
        int sidx = slice * 32 + q;
        s += h1[(size_t)b * H1P + t + sidx] * W2[t * KSZ + sidx];
    }
    red[tid] = s;
    __syncthreads();
    if (tid < 16) {
        float tot = 0.0f;
        #pragma unroll
        for (int sl = 0; sl < 16; ++sl) tot += red[sl * 16 + tid];
        h2[(size_t)tid * O2 + t] = tot + b2[t];
    }
}

// ---------------- kernel 3: gate g[b] = lin * sigmoid(sig) ----------------
__global__ __launch_bounds__(128) void k3_gate(const float* __restrict__ h2,
                                               const float* __restrict__ Wl,
                                               const float* __restrict__ bl,
                                               const float* __restrict__ Ws,
                                               const float* __restrict__ bs,
                                               float* __restrict__ g)
{
    __shared__ float rl[128], rs[128];
    int b = blockIdx.x, tid = threadIdx.x;
    float sl = 0.0f, ss = 0.0f;
    #pragma unroll
    for (int s = tid; s < O2; s += 128) {
        float v = h2[(size_t)b * O2 + s];
        sl += v * Wl[s];
        ss += v * Ws[s];
    }
    rl[tid] = sl; rs[tid] = ss;
    __syncthreads();
    for (int off = 64; off > 0; off >>= 1) {
        if (tid < off) { rl[tid] += rl[tid + off]; rs[tid] += rs[tid + off]; }
        __syncthreads();
    }
    if (tid == 0) {
        float lin = rl[0] + bl[0];
        float z   = rs[0] + bs[0];
        g[b] = lin * (1.0f / (1.0f + expf(-z)));
    }
}

// ---------------- kernel 4: residual + batch-norm ----------------
__global__ __launch_bounds__(256) void k4_out(const float* __restrict__ x,
                                              const float* __restrict__ g,
                                              const float* __restrict__ gamma,
                                              const float* __restrict__ beta,
                                              const float* __restrict__ mm,
                                              const float* __restrict__ mv,
                                              float* __restrict__ out)
{
    int i = blockIdx.x * 256 + threadIdx.x;
    if (i >= NB * LOUT * NC) return;
    int c = i & 15;
    int l = (i >> 4) % LOUT;
    int b = i / (LOUT * NC);
    float xv  = (l < LIN) ? x[((size_t)b * LIN + l) * NC + c] : 0.0f;
    float val = xv + g[b];
    out[i] = (val - mm[c]) * rsqrtf(mv[c] + EPSV) * gamma[c] + beta[c];
}

extern "C" void kernel_launch(void* const* d_in, const int* in_sizes, int n_in,
                              void* d_out, int out_size, void* d_ws, size_t ws_size,
                              hipStream_t stream)
{
    const float* x     = (const float*)d_in[0];
    const float* W1    = (const float*)d_in[1];
    const float* b1    = (const float*)d_in[2];
    const float* W2    = (const float*)d_in[3];
    const float* b2    = (const float*)d_in[4];
    const float* Wl    = (const float*)d_in[5];
    const float* bl    = (const float*)d_in[6];
    const float* Ws    = (const float*)d_in[7];
    const float* bs    = (const float*)d_in[8];
    const float* gamma = (const float*)d_in[9];
    const float* beta  = (const float*)d_in[10];
    const float* mm    = (const float*)d_in[11];
    const float* mv    = (const float*)d_in[12];

    float* h1 = (float*)d_ws;            // [16][1024] accumulator, then ELU'd h1
    float* h2 = h1 + NB * H1P;           // [16][512]
    float* g  = h2 + NB * O2;            // [16]

    k0_zero<<<(NB * H1P + 255) / 256, 256, 0, stream>>>(h1);
    k1_h1<<<64 * KSPLIT, 256, 0, stream>>>(x, W1, h1);
    k1b_elu<<<(NB * O1 + 255) / 256, 256, 0, stream>>>(h1, b1);
    k2_h2<<<O2, 256, 0, stream>>>(h1, W2, b2, h2);
    k3_gate<<<NB, 128, 0, stream>>>(h2, Wl, bl, Ws, bs, g);
    int total = NB * LOUT * NC;
    k4_out<<<(total + 255) / 256, 256, 0, stream>>>(x, g, gamma, beta, mm, mv,
                                                    (float*)d_out);
}